// KalmanFilter_21036749816100
// MI455X (gfx1250) — compile-verified
//
#include <hip/hip_runtime.h>

// Problem constants (from reference)
#define B_DIM 64
#define S_DIM 256
#define C_DIM 512
#define M_DIM (B_DIM * S_DIM)   // 16384 rows

// Tiling
#define BM 128
#define BN 64
#define BK 32
#define PITCH 36   // floats; 144B rows: 16B-aligned, bank-conflict-free 16-lane column reads
#define NSTAGE 2   // double-buffered LDS

typedef __attribute__((ext_vector_type(2))) float v2f;
typedef __attribute__((ext_vector_type(4))) float v4f;
typedef __attribute__((ext_vector_type(8))) float v8f;
typedef __attribute__((ext_vector_type(4))) int   v4i;

// gfx1250 async global->LDS copy path (ASYNCcnt), if the toolchain exposes it.
#if defined(__AMDGCN__) && __has_builtin(__builtin_amdgcn_global_load_async_to_lds_b128) && \
    __has_builtin(__builtin_amdgcn_s_wait_asynccnt)
#define USE_ASYNC_LDS 1
#else
#define USE_ASYNC_LDS 0
#endif

#if USE_ASYNC_LDS
__device__ __forceinline__ void async_copy16(const float* g, float* l) {
  // Param 1 is addrspace(1) v4i* (diagnosed as "__device__ int __vector(4) *"),
  // param 2 is the LDS destination.
  __builtin_amdgcn_global_load_async_to_lds_b128(
      (__attribute__((address_space(1))) v4i*)g,
      (__attribute__((address_space(3))) v4i*)l,
      /*offset=*/0, /*cpol=*/0);
}
#endif

// ---------------------------------------------------------------------------
// Kernel 0: collapse the Kalman scan to scalar gains.
// Q = q*I, H = h*I, P0 = I  =>  P stays p*I:
//   p_pred = p + q ; k = p_pred / (p_pred + h) ; p = (1-k) * p_pred
// kcoef[0] = 0 so the uniform blend formula yields states[:,0] = preds[:,0].
// ---------------------------------------------------------------------------
__global__ void kalman_coef_kernel(const float* __restrict__ Q,
                                   const float* __restrict__ H,
                                   float* __restrict__ kcoef) {
  if (threadIdx.x == 0 && blockIdx.x == 0) {
    const float q = Q[0];
    const float h = H[0];
    float p = 1.0f;
    kcoef[0] = 0.0f;
    for (int t = 1; t < S_DIM; ++t) {
      const float pp = p + q;
      const float k = pp / (pp + h);
      kcoef[t] = k;
      p = (1.0f - k) * pp;
    }
  }
}

// ---------------------------------------------------------------------------
// Kernel 1: fused GEMM (preds = x @ W^T + b) + Kalman blend epilogue.
// Grid: (M/BM, C/BN), 256 threads (8 wave32s). Each wave: 32x32 patch
// = 2x2 tiles of V_WMMA_F32_16X16X4_F32. LDS double-buffered.
// ---------------------------------------------------------------------------
__launch_bounds__(256)
__global__ void kalman_gemm_blend_kernel(const float* __restrict__ x,
                                         const float* __restrict__ W,
                                         const float* __restrict__ bias,
                                         const float* __restrict__ kcoef,
                                         float* __restrict__ out) {
  __shared__ float As[NSTAGE][BM][PITCH];  // As[.][row][k] = x[m0+row][k0+k]
  __shared__ float Bs[NSTAGE][BN][PITCH];  // Bs[.][n][k]   = W[n0+n][k0+k]

  const int tid  = threadIdx.x;
  const int lane = tid & 31;
  const int wave = tid >> 5;
  const int wm   = wave >> 1;              // 0..3 : 32-row strip within block
  const int wn   = wave & 1;               // 0..1 : 32-col strip within block

  const int m0 = blockIdx.x * BM;
  const int n0 = blockIdx.y * BN;

  const int rowLane = lane & 15;           // M/N index within a 16-wide tile
  const int kSel    = (lane >> 4) << 1;    // lanes 0-15 -> K{0,1}; 16-31 -> K{2,3}

  // Per-thread staging addresses (A: 4 x float4, B: 2 x float4 per stage).
  const float* aG[4];
  int aL[4];
#pragma unroll
  for (int r = 0; r < 4; ++r) {
    const int v   = tid + r * 256;
    const int row = v >> 3;                // 8 float4 per 32-float row
    const int c4  = (v & 7) << 2;
    aG[r] = x + (size_t)(m0 + row) * C_DIM + c4;
    aL[r] = row * PITCH + c4;
  }
  const float* bG[2];
  int bL[2];
#pragma unroll
  for (int r = 0; r < 2; ++r) {
    const int v  = tid + r * 256;
    const int n  = v >> 3;
    const int c4 = (v & 7) << 2;
    bG[r] = W + (size_t)(n0 + n) * C_DIM + c4;
    bL[r] = n * PITCH + c4;
  }

  v8f acc[2][2] = {};

  auto compute = [&](int buf) {
#pragma unroll
    for (int kk = 0; kk < BK; kk += 4) {
      const v2f a0 = *(const v2f*)&As[buf][wm * 32 + rowLane][kk + kSel];
      const v2f a1 = *(const v2f*)&As[buf][wm * 32 + 16 + rowLane][kk + kSel];
      const v2f b0 = *(const v2f*)&Bs[buf][wn * 32 + rowLane][kk + kSel];
      const v2f b1 = *(const v2f*)&Bs[buf][wn * 32 + 16 + rowLane][kk + kSel];

      acc[0][0] = __builtin_amdgcn_wmma_f32_16x16x4_f32(
          false, a0, false, b0, (short)0, acc[0][0], false, false);
      acc[0][1] = __builtin_amdgcn_wmma_f32_16x16x4_f32(
          false, a0, false, b1, (short)0, acc[0][1], false, false);
      acc[1][0] = __builtin_amdgcn_wmma_f32_16x16x4_f32(
          false, a1, false, b0, (short)0, acc[1][0], false, false);
      acc[1][1] = __builtin_amdgcn_wmma_f32_16x16x4_f32(
          false, a1, false, b1, (short)0, acc[1][1], false, false);
    }
  };

#if USE_ASYNC_LDS
  // --- async global->LDS staging: no VGPR round-trip, ASYNCcnt tracked ---
  auto stage = [&](int buf, int k0) {
    float* Ab = &As[buf][0][0];
    float* Bb = &Bs[buf][0][0];
#pragma unroll
    for (int r = 0; r < 4; ++r) async_copy16(aG[r] + k0, Ab + aL[r]);
#pragma unroll
    for (int r = 0; r < 2; ++r) async_copy16(bG[r] + k0, Bb + bL[r]);
  };

  stage(0, 0);
  __builtin_amdgcn_s_wait_asynccnt(0);
  __syncthreads();
  for (int k0 = 0, it = 0; k0 < C_DIM; k0 += BK, ++it) {
    const int buf = it & 1;
    const bool hasNext = (k0 + BK) < C_DIM;
    if (hasNext) stage(buf ^ 1, k0 + BK);   // overlap with WMMAs below
    compute(buf);
    if (hasNext) __builtin_amdgcn_s_wait_asynccnt(0);
    __syncthreads();
  }
#else
  // --- register-staged fallback: batch all 6 loads, one wait, then store ---
  auto issue = [&](int k0, v4f* pa, v4f* pb) {
#pragma unroll
    for (int r = 0; r < 4; ++r) pa[r] = *(const v4f*)(aG[r] + k0);
#pragma unroll
    for (int r = 0; r < 2; ++r) pb[r] = *(const v4f*)(bG[r] + k0);
  };
  auto commit = [&](int buf, const v4f* pa, const v4f* pb) {
#pragma unroll
    for (int r = 0; r < 4; ++r) *(v4f*)(&As[buf][0][0] + aL[r]) = pa[r];
#pragma unroll
    for (int r = 0; r < 2; ++r) *(v4f*)(&Bs[buf][0][0] + bL[r]) = pb[r];
  };

  {
    v4f pa[4], pb[2];
    issue(0, pa, pb);
    commit(0, pa, pb);
  }
  __syncthreads();
  for (int k0 = 0, it = 0; k0 < C_DIM; k0 += BK, ++it) {
    const int buf = it & 1;
    const bool hasNext = (k0 + BK) < C_DIM;
    v4f pa[4], pb[2];
    if (hasNext) issue(k0 + BK, pa, pb);    // loads in flight during WMMAs
    compute(buf);
    if (hasNext) commit(buf ^ 1, pa, pb);
    __syncthreads();
  }
#endif

  // ---- epilogue: bias + Kalman scalar blend ----
  // C/D layout: VGPR e -> M = e (lanes 0-15) or e+8 (lanes 16-31); N = lane&15.
  const int eRow = (lane >> 4) << 3;
#pragma unroll
  for (int i = 0; i < 2; ++i) {
#pragma unroll
    for (int j = 0; j < 2; ++j) {
      const int n    = n0 + wn * 32 + j * 16 + rowLane;
      const float bn = bias[n];
#pragma unroll
      for (int e = 0; e < 8; ++e) {
        const int m = m0 + wm * 32 + i * 16 + eRow + e;
        const float pred = acc[i][j][e] + bn;
        const float kv   = kcoef[m & (S_DIM - 1)];   // s = m % S_LEN
        const float xv   = x[(size_t)m * C_DIM + n];
        out[(size_t)m * C_DIM + n] = pred + kv * (xv - pred);
      }
    }
  }
}

// ---------------------------------------------------------------------------
extern "C" void kernel_launch(void* const* d_in, const int* in_sizes, int n_in,
                              void* d_out, int out_size, void* d_ws, size_t ws_size,
                              hipStream_t stream) {
  const float* x = (const float*)d_in[0];   // (64,256,512)
  const float* W = (const float*)d_in[1];   // (512,512)
  const float* b = (const float*)d_in[2];   // (512,)
  const float* Q = (const float*)d_in[3];   // (512,512) = q*I
  const float* H = (const float*)d_in[4];   // (512,512) = h*I
  float* out   = (float*)d_out;             // (64,256,512)
  float* kcoef = (float*)d_ws;              // 256 floats of scratch

  kalman_coef_kernel<<<1, 32, 0, stream>>>(Q, H, kcoef);

  dim3 grid(M_DIM / BM, C_DIM / BN);        // (128, 8)
  kalman_gemm_blend_kernel<<<grid, 256, 0, stream>>>(x, W, b, kcoef, out);
}